// SSIMLoss_34832184771315
// MI455X (gfx1250) — compile-verified
//
#include <hip/hip_runtime.h>

typedef __attribute__((ext_vector_type(2))) float v2f;
typedef __attribute__((ext_vector_type(8))) float v8f;

#define IMG_H     512
#define IMG_W     512
#define TILE      16
#define TILES_X   (IMG_W / TILE)   // 32
#define TILES_Y   (IMG_H / TILE)   // 32
#define PLANES    48               // 16 * 3
#define N_TILES   (TILES_X * TILES_Y * PLANES)  // 49152
#define RROWS     26               // 16 + 2*5 halo
#define RSTRIDE   28               // K padded to 28 (multiple of 4)
#define HSTRIDE   17               // H buffer row stride (bank-conflict-free)
#define HROWS     28               // 26 rows + 2 zero pad rows
#define HSZ       (HROWS * HSTRIDE)

// Normalized 1D Gaussian, window 11, sigma 1.5 (matches kornia/JAX fp32 values)
__device__ __constant__ float g_gauss[11] = {
    0.00102838f, 0.00759880f, 0.03600077f, 0.10936069f, 0.21300553f,
    0.26601180f, 0.21300553f, 0.10936069f, 0.03600077f, 0.00759880f,
    0.00102838f};

__device__ __forceinline__ v8f wmma_f32(v2f a, v2f b, v8f c) {
    // D = A(16x4,f32) * B(4x16,f32) + C(16x16,f32)
    return __builtin_amdgcn_wmma_f32_16x16x4_f32(
        /*neg_a=*/false, a, /*neg_b=*/false, b,
        /*c_mod=*/(short)0, c, /*reuse_a=*/false, /*reuse_b=*/false);
}

__global__ void __launch_bounds__(32)
ssim_tile_kernel(const float* __restrict__ img1,
                 const float* __restrict__ img2,
                 float* __restrict__ partial) {
    __shared__ float s_R1[RROWS * RSTRIDE];
    __shared__ float s_R2[RROWS * RSTRIDE];
    __shared__ float s_H [5 * HSZ];
    __shared__ float s_g [11];

    const int lane = threadIdx.x;            // wave32: one wave per block
    const int tx   = blockIdx.x;
    const int ty   = blockIdx.y;
    const int pl   = blockIdx.z;

    const size_t planeOff = (size_t)pl * (IMG_H * IMG_W);
    const int x0 = tx * TILE - 5;
    const int y0 = ty * TILE - 5;

    // hint the DMA path: prefetch the first halo rows of both images
    {
        int pr = min(max(y0, 0), IMG_H - 1);
        int pc = min(max(x0, 0), IMG_W - 1);
        __builtin_prefetch(img1 + planeOff + (size_t)pr * IMG_W + pc, 0, 3);
        __builtin_prefetch(img2 + planeOff + (size_t)pr * IMG_W + pc, 0, 3);
    }

    if (lane < 11) s_g[lane] = g_gauss[lane];

    // zero the H pad rows (26,27) of all 5 maps: 5*2*17 = 170 floats
    for (int i = lane; i < 5 * 2 * HSTRIDE; i += 32) {
        int m   = i / (2 * HSTRIDE);
        int rem = i % (2 * HSTRIDE);
        s_H[m * HSZ + (26 + rem / HSTRIDE) * HSTRIDE + (rem % HSTRIDE)] = 0.f;
    }

    // stage 26x28 halo regions (zero pad outside image & pad cols 26,27)
    for (int idx = lane; idx < RROWS * RSTRIDE; idx += 32) {
        int r = idx / RSTRIDE, c = idx % RSTRIDE;
        int gr = y0 + r, gc = x0 + c;
        bool ok = (c < 26) & (gr >= 0) & (gr < IMG_H) & (gc >= 0) & (gc < IMG_W);
        size_t off = planeOff + (size_t)gr * IMG_W + gc;
        s_R1[idx] = ok ? img1[off] : 0.f;
        s_R2[idx] = ok ? img2[off] : 0.f;
    }
    // single wave in the workgroup: DS ops are in-order, lockstep => no barrier

    const int Nn = lane & 15;               // N (or M) index of this lane
    const int ko = (lane >> 4) << 1;        // K sub-offset per lane half

    // Banded Gaussian fragments: B for horizontal pass == A for vertical pass
    v2f gfrag[7];
#pragma unroll
    for (int k7 = 0; k7 < 7; ++k7) {
        int K0 = k7 * 4 + ko;
        int d0 = K0 - Nn, d1 = K0 + 1 - Nn;
        float f0 = (d0 >= 0 && d0 <= 10) ? s_g[d0] : 0.f;
        float f1 = (d1 >= 0 && d1 <= 10) ? s_g[d1] : 0.f;
        gfrag[k7] = (v2f){f0, f1};
    }

    // ---- horizontal pass: H_m = R_m * G, maps m = {x, y, xx, yy, xy} ----
    v8f hAcc[10];
#pragma unroll
    for (int i = 0; i < 10; ++i)
        hAcc[i] = (v8f){0.f, 0.f, 0.f, 0.f, 0.f, 0.f, 0.f, 0.f};

#pragma unroll
    for (int k7 = 0; k7 < 7; ++k7) {
        int K0 = k7 * 4 + ko;
#pragma unroll
        for (int rbi = 0; rbi < 2; ++rbi) {       // row chunks 0..15, 10..25
            int row = rbi * 10 + Nn;
            float xa = s_R1[row * RSTRIDE + K0];
            float xb = s_R1[row * RSTRIDE + K0 + 1];
            float ya = s_R2[row * RSTRIDE + K0];
            float yb = s_R2[row * RSTRIDE + K0 + 1];
            v2f aX  = (v2f){xa, xb};
            v2f aY  = (v2f){ya, yb};
            v2f aXX = (v2f){xa * xa, xb * xb};
            v2f aYY = (v2f){ya * ya, yb * yb};
            v2f aXY = (v2f){xa * ya, xb * yb};
            hAcc[0 * 2 + rbi] = wmma_f32(aX,  gfrag[k7], hAcc[0 * 2 + rbi]);
            hAcc[1 * 2 + rbi] = wmma_f32(aY,  gfrag[k7], hAcc[1 * 2 + rbi]);
            hAcc[2 * 2 + rbi] = wmma_f32(aXX, gfrag[k7], hAcc[2 * 2 + rbi]);
            hAcc[3 * 2 + rbi] = wmma_f32(aYY, gfrag[k7], hAcc[3 * 2 + rbi]);
            hAcc[4 * 2 + rbi] = wmma_f32(aXY, gfrag[k7], hAcc[4 * 2 + rbi]);
        }
    }

    // spill H (C-layout: VGPR j -> row j + 8*(lane>=16), col = lane&15)
    const int mrow = (lane >> 4) << 3;
#pragma unroll
    for (int m = 0; m < 5; ++m) {
#pragma unroll
        for (int rbi = 0; rbi < 2; ++rbi) {
#pragma unroll
            for (int j = 0; j < 8; ++j) {
                int row = rbi * 10 + j + mrow;    // overlap rows identical
                s_H[m * HSZ + row * HSTRIDE + Nn] = hAcc[m * 2 + rbi][j];
            }
        }
    }

    // ---- vertical pass: O_m = G^T * H_m ----
    v8f oAcc[5];
#pragma unroll
    for (int m = 0; m < 5; ++m)
        oAcc[m] = (v8f){0.f, 0.f, 0.f, 0.f, 0.f, 0.f, 0.f, 0.f};

#pragma unroll
    for (int k7 = 0; k7 < 7; ++k7) {
        int K0 = k7 * 4 + ko;
#pragma unroll
        for (int m = 0; m < 5; ++m) {
            const float* Hm = s_H + m * HSZ;
            v2f b = (v2f){Hm[K0 * HSTRIDE + Nn], Hm[(K0 + 1) * HSTRIDE + Nn]};
            oAcc[m] = wmma_f32(gfrag[k7], b, oAcc[m]);
        }
    }

    // ---- pointwise SSIM + tile partial sum (layout-oblivious) ----
    const float c1 = 0.0001f;      // (0.01*1)^2
    const float c2 = 0.0009f;      // (0.03*1)^2
    const float eps = 1e-12f;
    float lsum = 0.f;
#pragma unroll
    for (int j = 0; j < 8; ++j) {
        float mu1 = oAcc[0][j], mu2 = oAcc[1][j];
        float bxx = oAcc[2][j], byy = oAcc[3][j], bxy = oAcc[4][j];
        float mu1s = mu1 * mu1, mu2s = mu2 * mu2, mu12 = mu1 * mu2;
        float s1 = bxx - mu1s, s2 = byy - mu2s, s12 = bxy - mu12;
        float num = (2.f * mu12 + c1) * (2.f * s12 + c2);
        float den = (mu1s + mu2s + c1) * (s1 + s2 + c2);
        float ssim = num / (den + eps);
        float loss = (1.f - ssim) * 0.5f;
        loss = fminf(fmaxf(loss, 0.f), 1.f);
        lsum += loss;
    }
#pragma unroll
    for (int off = 16; off > 0; off >>= 1)
        lsum += __shfl_down(lsum, off, 32);

    if (lane == 0) {
        int tileId = (pl * TILES_Y + ty) * TILES_X + tx;
        partial[tileId] = lsum;
    }
}

__global__ void __launch_bounds__(256)
ssim_reduce_kernel(const float* __restrict__ partial,
                   float* __restrict__ out, int n, float scale) {
    __shared__ float s[256];
    float t = 0.f;
    for (int i = threadIdx.x; i < n; i += 256) t += partial[i];  // fixed order
    s[threadIdx.x] = t;
    __syncthreads();
    for (int o = 128; o > 0; o >>= 1) {
        if ((int)threadIdx.x < o) s[threadIdx.x] += s[threadIdx.x + o];
        __syncthreads();
    }
    if (threadIdx.x == 0) out[0] = s[0] * scale;
}

extern "C" void kernel_launch(void* const* d_in, const int* in_sizes, int n_in,
                              void* d_out, int out_size, void* d_ws, size_t ws_size,
                              hipStream_t stream) {
    const float* img1 = (const float*)d_in[0];
    const float* img2 = (const float*)d_in[1];
    float* out     = (float*)d_out;
    float* partial = (float*)d_ws;   // 49152 floats = 192 KB scratch

    dim3 grid(TILES_X, TILES_Y, PLANES);
    ssim_tile_kernel<<<grid, 32, 0, stream>>>(img1, img2, partial);

    const float scale = 1.0f / (float)(16 * 3 * IMG_H * IMG_W);
    ssim_reduce_kernel<<<1, 256, 0, stream>>>(partial, out, N_TILES, scale);
}